// AMM_module_55027120996423
// MI455X (gfx1250) — compile-verified
//
#include <hip/hip_runtime.h>

// Problem constants (match reference: B=32, C_IN=3, H=W=512)
constexpr int IMGS = 32;
constexpr int CIN  = 3;
constexpr int Hh   = 512;
constexpr int Ww   = 512;

// Tile geometry: one 128-thread workgroup (4 wave32) -> 16 rows x 32 cols output
constexpr int TR    = 16;          // tile rows
constexpr int TC    = 32;          // tile cols
constexpr int PR    = TR + 2;      // staged patch rows (18)
constexpr int PSTR  = 40;          // staged patch cols / LDS row stride (floats)
                                   // cols c0-4 .. c0+35 (16B-aligned start)
constexpr int COLSH = 3;           // patch col of gc == c0-1  (halo shift)

typedef float v2f __attribute__((ext_vector_type(2)));
typedef float v8f __attribute__((ext_vector_type(8)));

#if defined(__has_builtin)
#if __has_builtin(__builtin_amdgcn_global_load_async_to_lds_b128) && \
    __has_builtin(__builtin_amdgcn_s_wait_asynccnt)
#define HAVE_ASYNC_LDS 1
#endif
#endif

#ifdef HAVE_ASYNC_LDS
// Builtin prototype (from hipcc diagnostic): first param is
// 'int __vector(4) __device__ *'  -> (global int4* src, shared int4* dst,
//                                     imm int offset, imm int cpol)
typedef int i32x4 __attribute__((vector_size(16)));
typedef __attribute__((address_space(1))) i32x4* gbl4_t;
typedef __attribute__((address_space(3))) i32x4* lds4_t;
#endif

__global__ __launch_bounds__(128)
void AMM_conv3x3_thresh_wmma(const float* __restrict__ x,
                             const float* __restrict__ w,
                             const float* __restrict__ bias,
                             float* __restrict__ out)
{
    __shared__ float patch[CIN * PR * PSTR];   // 3*18*40*4 = 8640 B

    const int tid = threadIdx.x;
    const int img = blockIdx.z;
    const int r0  = blockIdx.y * TR;
    const int c0  = blockIdx.x * TC;

    const float* xb = x + (size_t)img * (CIN * Hh * Ww);

    // Block-uniform interior test: whole 18x40 halo patch lies inside the image.
    const bool interior = (r0 >= 1) && (r0 + TR + 1 <= Hh) &&
                          (c0 >= 4) && (c0 + 36 <= Ww);

#ifdef HAVE_ASYNC_LDS
    if (interior) {
        // -------- async DMA staging: 16B per lane per op, ASYNCcnt-tracked ---
        // 3 ch * 18 rows * 10 chunks of 4 floats = 540 b128 transfers
        constexpr int NCHUNK = CIN * PR * (PSTR / 4);   // 540
        for (int it = 0; it < 5; ++it) {
            int idx = tid + it * 128;
            if (idx < NCHUNK) {
                int ch  = idx / (PR * (PSTR / 4));
                int rem = idx % (PR * (PSTR / 4));
                int rr  = rem / (PSTR / 4);
                int c4  = rem % (PSTR / 4);
                const float* g = xb + (size_t)ch * (Hh * Ww)
                                    + (size_t)(r0 - 1 + rr) * Ww
                                    + (c0 - 4) + c4 * 4;
                float* l = &patch[ch * (PR * PSTR) + rr * PSTR + c4 * 4];
                __builtin_amdgcn_global_load_async_to_lds_b128(
                    (gbl4_t)const_cast<float*>(g), (lds4_t)l, 0, 0);
            }
        }
        __builtin_amdgcn_s_wait_asynccnt(0);
    } else
#endif
    {
        // -------- scalar staging with zero 'same'-padding ---------------------
        for (int i = tid; i < CIN * PR * PSTR; i += 128) {
            int cc = i % PSTR;
            int t  = i / PSTR;
            int rr = t % PR;
            int ch = t / PR;
            int gr = r0 + rr - 1;
            int gc = c0 - 4 + cc;
            float v = 0.0f;
            if (gr >= 0 && gr < Hh && gc >= 0 && gc < Ww)
                v = xb[(size_t)ch * (Hh * Ww) + (size_t)gr * Ww + gc];
            patch[i] = v;
        }
    }
    __syncthreads();

    // ---------------- per-lane constants ------------------------------------
    const int lane = tid & 31;
    const int n    = lane & 15;    // matrix column N == pixel within strip
    const int h    = lane >> 4;    // lane half (K split for f32 16x16x4 layout)

    // 27 conv taps, padded to 28 (tap 27 weight == 0 kills its B garbage)
    float wv[28];
    #pragma unroll
    for (int k = 0; k < 27; ++k) wv[k] = w[k];
    wv[27] = 0.0f;
    const float bval = bias[0];

    // A fragments: weights replicated across all 16 M rows.
    // f32 16x4 A layout: VGPR0 -> K = 2h, VGPR1 -> K = 2h+1 (h = lane half)
    v2f afrag[7];
    #pragma unroll
    for (int t = 0; t < 7; ++t) {
        afrag[t][0] = wv[4 * t + 2 * h + 0];
        afrag[t][1] = wv[4 * t + 2 * h + 1];
    }

    // Per-chunk LDS float-offset bases for the B (im2col) operand.
    // tap k -> (cin, kh, kw); offset relative to patch[0][rr][col of pixel n]
    int boff0[7], boff1[7];
    #pragma unroll
    for (int t = 0; t < 7; ++t) {
        int k0 = 4 * t + 2 * h;            // K index held in B VGPR0
        int k1 = k0 + 1;                   // K index held in B VGPR1
        if (k0 > 26) k0 = 26;              // tap 27: address clamped, A-side zero
        if (k1 > 26) k1 = 26;
        int ci0 = k0 / 9, kh0 = (k0 % 9) / 3, kw0 = k0 % 3;
        int ci1 = k1 / 9, kh1 = (k1 % 9) / 3, kw1 = k1 % 3;
        boff0[t] = ci0 * (PR * PSTR) + kh0 * PSTR + kw0;
        boff1[t] = ci1 * (PR * PSTR) + kh1 * PSTR + kw1;
    }

    // ---------------- compute: 2 interleaved 16-pixel strips per wave iter --
    const int wave = tid >> 5;
    float* ob = out + (size_t)img * (Hh * Ww);

    for (int pr = wave; pr < TR; pr += 4) {      // uniform per wave: EXEC all-1s
        const int base0 = pr * PSTR + COLSH + n;       // strip 0: cols [0,16)
        const int base1 = pr * PSTR + COLSH + 16 + n;  // strip 1: cols [16,32)
        v8f acc0 = {};
        v8f acc1 = {};
        #pragma unroll
        for (int t = 0; t < 7; ++t) {
            v2f b0, b1;
            b0[0] = patch[base0 + boff0[t]];
            b0[1] = patch[base0 + boff1[t]];
            b1[0] = patch[base1 + boff0[t]];
            b1[1] = patch[base1 + boff1[t]];
            // D = A(weights) x B(im2col) + C, full fp32
            acc0 = __builtin_amdgcn_wmma_f32_16x16x4_f32(
                       false, afrag[t], false, b0, (short)0, acc0, false, false);
            acc1 = __builtin_amdgcn_wmma_f32_16x16x4_f32(
                       false, afrag[t], false, b1, (short)0, acc1, false, false);
        }
        // All D rows identical -> D VGPR0, lane L = conv(out pixel N=L%16).
        float c1a = acc0[0] + bval;              // conv + bias (strip 0)
        float c1b = acc1[0] + bval;              // conv + bias (strip 1)
        // softmax over 1 channel == 1.0; s1 = c1 + 1; hi = (s1 < 0.5) ? 0 : 1
        float v   = h ? c1b : c1a;               // lane half selects strip
        float res = (v < -0.5f) ? 0.0f : 1.0f;
        // 32 consecutive pixels per wave -> one coalesced 128B store
        ob[(size_t)(r0 + pr) * Ww + (c0 + lane)] = res;
    }
}

extern "C" void kernel_launch(void* const* d_in, const int* in_sizes, int n_in,
                              void* d_out, int out_size, void* d_ws, size_t ws_size,
                              hipStream_t stream) {
    const float* x  = (const float*)d_in[0];   // [32,3,512,512] f32
    const float* w  = (const float*)d_in[1];   // [1,3,3,3] f32 (27)
    const float* b  = (const float*)d_in[2];   // [1] f32
    float* out      = (float*)d_out;           // [32,1,512,512] f32

    dim3 grid(Ww / TC, Hh / TR, IMGS);         // 16 x 32 x 32 workgroups
    AMM_conv3x3_thresh_wmma<<<grid, 128, 0, stream>>>(x, w, b, out);
}